// RegionProposalNetwork_91190745629259
// MI455X (gfx1250) — compile-verified
//
#include <hip/hip_runtime.h>
#include <stdint.h>

// ---------------------------------------------------------------------------
// RPN proposal generation for MI455X (gfx1250, wave32).
// Pipeline: build sortable keys -> hierarchical bitonic sort (LDS chunk sort +
// global merge steps) -> gather+decode top-2000 -> 16x16-tile IoU bitmask
// (WMMA f32 16x16x4 computes the area_i+area_j union term; divide-free
// threshold test) -> one-wave serial NMS scan per batch (top-2000 packet
// staged into LDS by the Tensor Data Mover).
// ---------------------------------------------------------------------------

#define NB   8
#define AA   3
#define HH   200
#define WW   304
#define WA   (WW * AA)          // 912
#define HWA  (HH * WW * AA)     // 182400
#define PLANE (HH * WW)         // 60800
#define NPAD (1 << 18)          // 262144 (pad for bitonic sort)
#define CHUNK 2048              // elements sorted per LDS chunk (16 KB of u64)
#define PRE  2000
#define POST 1000
#define TILES 125               // 2000 / 16
#define IMGW 1216.0f
#define IMGH 800.0f
#define BBOX_CLIP 4.135166556742356f   // log(1000/16)
#define NMS_T 0.7f
#define PKT  12000              // per-batch packet: 8000 box + 2000 prob + 2000 valid

typedef __attribute__((ext_vector_type(2))) float    v2f;
typedef __attribute__((ext_vector_type(8))) float    v8f;
typedef __attribute__((ext_vector_type(4))) unsigned v4u;
typedef __attribute__((ext_vector_type(4))) int      v4i;
typedef __attribute__((ext_vector_type(8))) int      v8i;

#if defined(__has_builtin)
#if __has_builtin(__builtin_amdgcn_wmma_f32_16x16x4_f32)
#define HAVE_WMMA4 1
#endif
#if __has_builtin(__builtin_amdgcn_tensor_load_to_lds) && __has_builtin(__builtin_amdgcn_s_wait_tensorcnt)
#define HAVE_TDM 1
#endif
#endif

// Monotonic float -> u32 key (descending float == descending key).
__device__ __forceinline__ unsigned f2key(float f) {
  unsigned u = __float_as_uint(f);
  return (u & 0x80000000u) ? ~u : (u | 0x80000000u);
}

// ---------------------------------------------------------------------------
// K0: zero the (8,1000,5) output (harness poisons it).
// ---------------------------------------------------------------------------
__global__ void k_zero(float* __restrict__ out, int n) {
  int i = blockIdx.x * blockDim.x + threadIdx.x;
  if (i < n) out[i] = 0.0f;
}

// ---------------------------------------------------------------------------
// K1: per-(batch, anchor) composite sort key.
//   negcomp = (~key(score) << 32) | j   -> ascending sort == score-descending,
//   ties broken toward smaller anchor index (matches stable top_k/argsort).
//   Padding slots get ~0ull (sort last).
// ---------------------------------------------------------------------------
__global__ void k_build_keys(const float* __restrict__ obj,
                             unsigned long long* __restrict__ keys) {
  unsigned gid = blockIdx.x * blockDim.x + threadIdx.x;
  unsigned n = gid >> 18;            // / NPAD
  unsigned p = gid & (NPAD - 1);
  unsigned long long v = ~0ull;
  if (p < HWA) {
    int h = (int)(p / WA);
    int rem = (int)(p % WA);
    int w = rem / AA;
    int a = rem % AA;
    float s = obj[(((size_t)n * AA + a) * HH + h) * WW + w];
    unsigned key = f2key(s);
    v = ((unsigned long long)(~key) << 32) | (unsigned long long)p;
  }
  keys[gid] = v;
}

// ---------------------------------------------------------------------------
// K2a: full bitonic sort of each 2048-element chunk in LDS (stages k=2..2048).
// Direction flags use the flat index, which equals the within-batch index
// modulo NPAD (batch bases are multiples of NPAD, so & k is unaffected).
// ---------------------------------------------------------------------------
__global__ __launch_bounds__(256) void k_sort_local(unsigned long long* __restrict__ keys) {
  __shared__ unsigned long long s[CHUNK];
  size_t gbase = (size_t)blockIdx.x * CHUNK;
  for (int t = threadIdx.x; t < CHUNK; t += 256) s[t] = keys[gbase + t];
  __syncthreads();
  for (unsigned k = 2; k <= CHUNK; k <<= 1) {
    for (unsigned j = k >> 1; j > 0; j >>= 1) {
      for (int t = threadIdx.x; t < CHUNK; t += 256) {
        unsigned i = (unsigned)t;
        unsigned ixj = i ^ j;
        if (ixj > i) {
          bool asc = (((gbase + i) & (size_t)k) == 0);
          unsigned long long x = s[i], y = s[ixj];
          if (asc ? (x > y) : (x < y)) { s[i] = y; s[ixj] = x; }
        }
      }
      __syncthreads();
    }
  }
  for (int t = threadIdx.x; t < CHUNK; t += 256) keys[gbase + t] = s[t];
}

// ---------------------------------------------------------------------------
// K2b: one global compare-exchange step (only needed while j >= CHUNK).
// ---------------------------------------------------------------------------
__global__ void k_bitonic_global(unsigned long long* __restrict__ keys,
                                 unsigned k, unsigned j) {
  unsigned gid = blockIdx.x * blockDim.x + threadIdx.x;
  unsigned n = gid >> 18;
  unsigned i = gid & (NPAD - 1);
  unsigned ixj = i ^ j;
  if (ixj > i) {
    unsigned long long* b = keys + (size_t)n * NPAD;
    unsigned long long x = b[i], y = b[ixj];
    bool asc = ((i & k) == 0);
    if (asc ? (x > y) : (x < y)) { b[i] = y; b[ixj] = x; }
  }
}

// ---------------------------------------------------------------------------
// K2c: finish a merge stage k in LDS (steps j = CHUNK/2 .. 1).
// ---------------------------------------------------------------------------
__global__ __launch_bounds__(256) void k_merge_local(unsigned long long* __restrict__ keys,
                                                     unsigned k) {
  __shared__ unsigned long long s[CHUNK];
  size_t gbase = (size_t)blockIdx.x * CHUNK;
  for (int t = threadIdx.x; t < CHUNK; t += 256) s[t] = keys[gbase + t];
  __syncthreads();
  for (unsigned j = CHUNK >> 1; j > 0; j >>= 1) {
    for (int t = threadIdx.x; t < CHUNK; t += 256) {
      unsigned i = (unsigned)t;
      unsigned ixj = i ^ j;
      if (ixj > i) {
        bool asc = (((gbase + i) & (size_t)k) == 0);
        unsigned long long x = s[i], y = s[ixj];
        if (asc ? (x > y) : (x < y)) { s[i] = y; s[ixj] = x; }
      }
    }
    __syncthreads();
  }
  for (int t = threadIdx.x; t < CHUNK; t += 256) keys[gbase + t] = s[t];
}

// ---------------------------------------------------------------------------
// K3: gather top-2000 per batch, decode vs anchors, clip, sigmoid, validity.
// Packet layout per batch: [0,8000) boxes, [8000,10000) probs, [10000,12000) valid.
// ---------------------------------------------------------------------------
__global__ void k_gather_decode(const float* __restrict__ obj,
                                const float* __restrict__ del,
                                const float* __restrict__ anc,
                                const unsigned long long* __restrict__ keys,
                                float* __restrict__ topk) {
  int tid = blockIdx.x * blockDim.x + threadIdx.x;
  if (tid >= NB * PRE) return;
  int n = tid / PRE, r = tid % PRE;

  unsigned long long v = keys[(size_t)n * NPAD + r];
  unsigned j = (unsigned)(v & 0xFFFFFFFFull);
  int h = (int)(j / WA);
  int rem = (int)(j % WA);
  int w = rem / AA;
  int a = rem % AA;
  size_t hw = (size_t)h * WW + w;

  float score = obj[(((size_t)n * AA + a) * HH + h) * WW + w];

  size_t dbase = ((size_t)n * 4 * AA + a * 4) * PLANE + hw;
  float dx = del[dbase];
  float dy = del[dbase + PLANE];
  float dw = fminf(del[dbase + 2 * (size_t)PLANE], BBOX_CLIP);
  float dh = fminf(del[dbase + 3 * (size_t)PLANE], BBOX_CLIP);

  float ax1 = anc[(size_t)j * 4 + 0], ay1 = anc[(size_t)j * 4 + 1];
  float ax2 = anc[(size_t)j * 4 + 2], ay2 = anc[(size_t)j * 4 + 3];
  float wa = ax2 - ax1, ha = ay2 - ay1;
  float cx = ax1 + 0.5f * wa, cy = ay1 + 0.5f * ha;
  float pcx = dx * wa + cx, pcy = dy * ha + cy;
  float pw = expf(dw) * wa, ph = expf(dh) * ha;

  float x1 = fminf(fmaxf(pcx - 0.5f * pw, 0.0f), IMGW);
  float y1 = fminf(fmaxf(pcy - 0.5f * ph, 0.0f), IMGH);
  float x2 = fminf(fmaxf(pcx + 0.5f * pw, 0.0f), IMGW);
  float y2 = fminf(fmaxf(pcy + 0.5f * ph, 0.0f), IMGH);

  float prob = 1.0f / (1.0f + expf(-score));
  float valid = ((x2 - x1) >= 1.0f && (y2 - y1) >= 1.0f && prob >= 0.0f) ? 1.0f : 0.0f;

  float* tb = topk + (size_t)n * PKT;
  tb[r * 4 + 0] = x1;
  tb[r * 4 + 1] = y1;
  tb[r * 4 + 2] = x2;
  tb[r * 4 + 3] = y2;
  tb[8000 + r] = prob;
  tb[10000 + r] = valid;
}

// ---------------------------------------------------------------------------
// K4: 16x16 IoU tile per wave32.  Union term area_i + area_j is computed as a
// rank-2 outer product on the WMMA unit (V_WMMA_F32_16X16X4_F32):
//   A(16x4) row M = [area_row(M), 1, 0, 0];  B(4x16) col N = [1, area_col(N), 0, 0]^T
// C/D layout: lane = 16*(M/8) + N, VGPR v = M%8 -> per-VGPR ballot packs one
// 16-bit mask word per row per column-tile.
// Threshold test is divide-free: inter > T*(union) with union = sum - inter.
// (union >= max(area) >= 0; union==0 => inter==0 => false, matching NaN>T.)
// ---------------------------------------------------------------------------
__global__ __launch_bounds__(32) void k_iou_mask(const float* __restrict__ topk,
                                                 unsigned* __restrict__ mask) {
  int n = blockIdx.z, ti = blockIdx.y, tj = blockIdx.x;
  int l = threadIdx.x;

  __shared__ float rb[16][4], cb[16][4];
  __shared__ float rar[16], car[16];

  const float* tb = topk + (size_t)n * PKT;
  if (l < 16) {
    int rr = ti * 16 + l;
    #pragma unroll
    for (int k = 0; k < 4; ++k) rb[l][k] = tb[rr * 4 + k];
    rar[l] = (rb[l][2] - rb[l][0]) * (rb[l][3] - rb[l][1]);
  } else {
    int cc = tj * 16 + (l - 16);
    #pragma unroll
    for (int k = 0; k < 4; ++k) cb[l - 16][k] = tb[cc * 4 + k];
    car[l - 16] = (cb[l - 16][2] - cb[l - 16][0]) * (cb[l - 16][3] - cb[l - 16][1]);
  }
  __syncthreads();

  int col = l & 15;
  int rbase = (l >> 4) * 8;

  float uni[8];
#if defined(HAVE_WMMA4)
  {
    // A: lanes 0-15 hold (M=lane, K=0..1); lanes 16-31 hold (M=lane-16, K=2..3).
    // B: lanes 0-15 hold (K=0..1, N=lane); lanes 16-31 hold (K=2..3, N=lane-16).
    v2f a = {0.0f, 0.0f}, b = {0.0f, 0.0f};
    if (l < 16) { a.x = rar[l]; a.y = 1.0f; b.x = 1.0f; b.y = car[l]; }
    v8f c = {0.0f, 0.0f, 0.0f, 0.0f, 0.0f, 0.0f, 0.0f, 0.0f};
    c = __builtin_amdgcn_wmma_f32_16x16x4_f32(
        /*neg_a=*/false, a, /*neg_b=*/false, b,
        /*c_mod=*/(short)0, c, /*reuse_a=*/false, /*reuse_b=*/false);
    #pragma unroll
    for (int v = 0; v < 8; ++v) uni[v] = c[v];
  }
#else
  #pragma unroll
  for (int v = 0; v < 8; ++v) uni[v] = rar[rbase + v] + car[col];
#endif

  #pragma unroll
  for (int v = 0; v < 8; ++v) {
    int rr = rbase + v;
    float lx = fmaxf(rb[rr][0], cb[col][0]);
    float ly = fmaxf(rb[rr][1], cb[col][1]);
    float rx = fminf(rb[rr][2], cb[col][2]);
    float ry = fminf(rb[rr][3], cb[col][3]);
    float iw = fmaxf(rx - lx, 0.0f);
    float ih = fmaxf(ry - ly, 0.0f);
    float inter = iw * ih;
    bool over = inter > NMS_T * (uni[v] - inter);   // divide-free IoU > 0.7
    unsigned long long bal = __ballot(over);
    if (l == 0) {
      int rowlo = ti * 16 + v;
      int rowhi = rowlo + 8;
      mask[((size_t)n * PRE + rowlo) * TILES + tj] = (unsigned)(bal & 0xFFFFull);
      mask[((size_t)n * PRE + rowhi) * TILES + tj] = (unsigned)((bal >> 16) & 0xFFFFull);
    }
  }
}

// ---------------------------------------------------------------------------
// K5: one wave32 per batch.  Stage the 48 KB top-2000 packet into LDS via the
// Tensor Data Mover, then do the serial greedy suppression scan and scatter
// kept (box, prob) rows into the output.
// ---------------------------------------------------------------------------
__global__ __launch_bounds__(32) void k_nms_scan(const float* __restrict__ topk,
                                                 const unsigned* __restrict__ mask,
                                                 float* __restrict__ out) {
  int n = blockIdx.x;
  int l = threadIdx.x;

  __shared__ float sm[PKT];            // boxes | probs | valid
  __shared__ unsigned removed[TILES];

#if defined(HAVE_TDM)
  {
    unsigned long long ga = (unsigned long long)(uintptr_t)(topk + (size_t)n * PKT);
    unsigned ldsoff = (unsigned)(uintptr_t)(void*)sm;   // AS3 offset in low 32 bits
    // D# group0: count=1, lds_addr, global_addr[56:0], type=2.
    v4u g0;
    g0.x = 1u;
    g0.y = ldsoff;
    g0.z = (unsigned)(ga & 0xFFFFFFFFull);
    g0.w = (unsigned)((ga >> 32) & 0x01FFFFFFull) | 0x80000000u;
    // D# group1: data_size=4B, tensor_dim0=PKT, tensor_dim1=1,
    //            tile_dim0=PKT, tile_dim1=1, stride0=PKT.
    v8i g1;
    g1[0] = 0x00020000;
    g1[1] = (int)(((unsigned)PKT & 0xFFFFu) << 16);
    g1[2] = (int)(((unsigned)PKT >> 16) | (1u << 16));
    g1[3] = (int)((unsigned)PKT << 16);
    g1[4] = 1;
    g1[5] = PKT;
    g1[6] = 0;
    g1[7] = 0;
    v4i g2 = {1, 0, 0, 0};     // tensor_dim2 = 1, rest unused
    v4i g3 = {0, 0, 0, 0};
#if __has_include(<hip/amd_detail/amd_gfx1250_TDM.h>)
    v8i g4 = {0, 0, 0, 0, 0, 0, 0, 0};
    __builtin_amdgcn_tensor_load_to_lds(g0, g1, g2, g3, g4, 0);
#else
    __builtin_amdgcn_tensor_load_to_lds(g0, g1, g2, g3, 0);
#endif
    __builtin_amdgcn_s_wait_tensorcnt(0);
  }
#else
  for (int i = l; i < PKT; i += 32) sm[i] = topk[(size_t)n * PKT + i];
#endif

  for (int i = l; i < TILES; i += 32) removed[i] = 0u;
  __syncthreads();

  int kept = 0;
  for (int i = 0; i < PRE; ++i) {
    bool rm = (removed[i >> 4] >> (i & 15)) & 1u;
    bool keep = (!rm) && (sm[10000 + i] > 0.5f);
    if (keep) {
      const unsigned* mrow = mask + ((size_t)n * PRE + i) * TILES;
      for (int wi = l; wi < TILES; wi += 32) removed[wi] |= mrow[wi];
      if (kept < POST) {
        size_t o = ((size_t)n * POST + kept) * 5;
        if (l < 4) out[o + l] = sm[i * 4 + l];
        if (l == 4) out[o + 4] = sm[8000 + i];
      }
      ++kept;
    }
    __syncthreads();
  }
}

// ---------------------------------------------------------------------------
// Host launcher.  Workspace layout (bytes):
//   [0, 16777216)            sort keys   : 8 * 262144 * u64
//   [16777216, 17161216)     top-k packet: 8 * 12000 * f32
//   [17161216, 25161216)     IoU bitmask : 8 * 2000 * 125 * u32
// ---------------------------------------------------------------------------
extern "C" void kernel_launch(void* const* d_in, const int* in_sizes, int n_in,
                              void* d_out, int out_size, void* d_ws, size_t ws_size,
                              hipStream_t stream) {
  const float* obj = (const float*)d_in[0];
  const float* del = (const float*)d_in[1];
  const float* anc = (const float*)d_in[2];
  float* out = (float*)d_out;

  char* base = (char*)d_ws;
  unsigned long long* keys = (unsigned long long*)base;
  float* topk = (float*)(base + (size_t)NB * NPAD * 8);
  unsigned* mask = (unsigned*)(base + (size_t)NB * NPAD * 8 + (size_t)NB * PKT * 4);

  k_zero<<<(NB * POST * 5 + 255) / 256, 256, 0, stream>>>(out, NB * POST * 5);

  k_build_keys<<<(NB * NPAD) / 256, 256, 0, stream>>>(obj, keys);

  // Hierarchical bitonic sort: LDS chunk sort, then per-stage global steps
  // (j >= CHUNK) finished by an in-LDS merge of the low steps.
  k_sort_local<<<(NB * NPAD) / CHUNK, 256, 0, stream>>>(keys);
  for (unsigned k = CHUNK * 2; k <= (unsigned)NPAD; k <<= 1) {
    for (unsigned j = k >> 1; j >= CHUNK; j >>= 1)
      k_bitonic_global<<<(NB * NPAD) / 256, 256, 0, stream>>>(keys, k, j);
    k_merge_local<<<(NB * NPAD) / CHUNK, 256, 0, stream>>>(keys, k);
  }

  k_gather_decode<<<(NB * PRE + 255) / 256, 256, 0, stream>>>(obj, del, anc, keys, topk);

  k_iou_mask<<<dim3(TILES, TILES, NB), 32, 0, stream>>>(topk, mask);

  k_nms_scan<<<NB, 32, 0, stream>>>(topk, mask, out);

  (void)in_sizes; (void)n_in; (void)out_size; (void)ws_size;
}